// ContractProduct3j_45071386804928
// MI455X (gfx1250) — compile-verified
//
#include <hip/hip_runtime.h>
#include <hip/hip_bf16.h>

typedef float v2f __attribute__((ext_vector_type(2)));
typedef float v8f __attribute__((ext_vector_type(8)));

// ---------------------------------------------------------------------------
// Geometry of the problem
//   inputs : [NC][16] fp32 (l = 0..3 blocks at offsets l*l)
//   output : [NC][25] fp32 (l = 0..4 blocks at offsets l*l)
//   out[s] = sum_{i,j} t1[i] * t2[j] * W[i][j][s]
// B table stored s-major, padded: Bt[s][k], s in [0,32), k in [0,256),
// row stride 260 floats (bank-conflict-free b64 reads from LDS).
// ---------------------------------------------------------------------------
#define B_STRIDE 260
#define B_FLOATS (32 * B_STRIDE)

// 27 (l1,l2,l3) triples with l1,l2<=3, |l1-l2|<=l3<=min(l1+l2,4), even parity
static __device__ const int TRIPLES[27][3] = {
    {0,0,0},
    {0,1,1},{0,2,2},{0,3,3},
    {1,0,1},{1,1,0},{1,1,2},{1,2,1},{1,2,3},{1,3,2},{1,3,4},
    {2,0,2},{2,1,1},{2,1,3},{2,2,0},{2,2,2},{2,2,4},{2,3,1},{2,3,3},
    {3,0,3},{3,1,2},{3,1,4},{3,2,1},{3,2,3},{3,3,0},{3,3,2},{3,3,4}
};

// ------------------------- Wigner-3j generation ----------------------------
struct cplx { double re, im; };

__device__ inline cplx cadd(cplx a, cplx b) { return {a.re + b.re, a.im + b.im}; }
__device__ inline cplx cmul(cplx a, cplx b) {
    return {a.re * b.re - a.im * b.im, a.re * b.im + a.im * b.re};
}
__device__ inline cplx cscale(cplx a, double s) { return {a.re * s, a.im * s}; }

__device__ inline double dfact(int n) {
    double r = 1.0;
    for (int i = 2; i <= n; ++i) r *= (double)i;
    return r;
}

__device__ double su2_cg(int j1, int m1, int j2, int m2, int j3, int m3) {
    if (m3 != m1 + m2) return 0.0;
    int vmin = max(max(-j1 + j2 + m3, -j1 + m1), 0);
    int vmax = min(min(j2 + j3 + m1, j3 - j1 + j2), j3 + m3);
    if (vmax < vmin) return 0.0;
    double C = sqrt((2.0 * j3 + 1.0) * dfact(j3 + j1 - j2) * dfact(j3 - j1 + j2) *
                    dfact(j1 + j2 - j3) * dfact(j3 + m3) * dfact(j3 - m3) /
                    (dfact(j1 + j2 + j3 + 1) * dfact(j1 - m1) * dfact(j1 + m1) *
                     dfact(j2 - m2) * dfact(j2 + m2)));
    double S = 0.0;
    for (int v = vmin; v <= vmax; ++v) {
        double sgn = ((v + j2 + m2) & 1) ? -1.0 : 1.0;
        S += sgn * dfact(j2 + j3 + m1 - v) * dfact(j1 - m1 + v) /
             (dfact(v) * dfact(j3 - j1 + j2 - v) * dfact(j3 + m3 - v) *
              dfact(j1 - j2 - m3 + v));
    }
    return C * S;
}

// real->complex change-of-basis matrix entry q[a][b] for degree l (incl. (-i)^l)
__device__ cplx qentry(int l, int a, int b) {
    const double s = 0.70710678118654752440;
    int m = a - l;
    double re = 0.0, im = 0.0;
    if (m < 0) {
        if (b == l - m) re = s;
        else if (b == l + m) im = -s;
    } else if (m == 0) {
        if (b == l) re = 1.0;
    } else {
        double sg = (m & 1) ? -1.0 : 1.0;
        if (b == l + m) re = sg * s;
        else if (b == l - m) im = sg * s;
    }
    switch (l & 3) {                 // multiply by (-i)^l
        case 0:  return { re,  im};
        case 1:  return { im, -re};
        case 2:  return {-re, -im};
        default: return {-im,  re};
    }
}

// one block (64 threads) per triple; fills the padded B table in d_ws
__global__ void w3j_gen_kernel(float* __restrict__ Bt) {
    __shared__ double CG[441];
    __shared__ double R[441];
    __shared__ cplx  QA[49], QB[49], QC[81];
    __shared__ cplx  TA[441], TB[441];
    __shared__ double red[64];

    const int tid = threadIdx.x;
    const int l1 = TRIPLES[blockIdx.x][0];
    const int l2 = TRIPLES[blockIdx.x][1];
    const int l3 = TRIPLES[blockIdx.x][2];
    const int n1 = 2 * l1 + 1, n2 = 2 * l2 + 1, n3 = 2 * l3 + 1;
    const int tot = n1 * n2 * n3;

    for (int idx = tid; idx < n1 * n1; idx += 64) QA[idx] = qentry(l1, idx / n1, idx % n1);
    for (int idx = tid; idx < n2 * n2; idx += 64) QB[idx] = qentry(l2, idx / n2, idx % n2);
    for (int idx = tid; idx < n3 * n3; idx += 64) QC[idx] = qentry(l3, idx / n3, idx % n3);
    for (int idx = tid; idx < tot; idx += 64) {
        int i = idx / (n2 * n3), k = (idx / n3) % n2, m = idx % n3;
        CG[idx] = su2_cg(l1, i - l1, l2, k - l2, l3, m - l3);
    }
    __syncthreads();

    // TA[j,k,m] = sum_i QA[i,j] * CG[i,k,m]
    for (int idx = tid; idx < tot; idx += 64) {
        int j = idx / (n2 * n3), k = (idx / n3) % n2, m = idx % n3;
        cplx acc = {0.0, 0.0};
        for (int i = 0; i < n1; ++i)
            acc = cadd(acc, cscale(QA[i * n1 + j], CG[(i * n2 + k) * n3 + m]));
        TA[idx] = acc;
    }
    __syncthreads();

    // TB[j,l,m] = sum_k QB[k,l] * TA[j,k,m]
    for (int idx = tid; idx < tot; idx += 64) {
        int j = idx / (n2 * n3), l = (idx / n3) % n2, m = idx % n3;
        cplx acc = {0.0, 0.0};
        for (int k = 0; k < n2; ++k)
            acc = cadd(acc, cmul(QB[k * n2 + l], TA[(j * n2 + k) * n3 + m]));
        TB[idx] = acc;
    }
    __syncthreads();

    // R[j,l,n] = Re( sum_m QC[m,n] * TB[j,l,m] )
    for (int idx = tid; idx < tot; idx += 64) {
        int j = idx / (n2 * n3), l = (idx / n3) % n2, n = idx % n3;
        double acc = 0.0;
        for (int m = 0; m < n3; ++m) {
            cplx t = TB[(j * n2 + l) * n3 + m];
            cplx q = QC[m * n3 + n];
            acc += q.re * t.re - q.im * t.im;
        }
        R[idx] = acc;
    }
    __syncthreads();

    // Frobenius norm
    double s = 0.0;
    for (int idx = tid; idx < tot; idx += 64) s += R[idx] * R[idx];
    red[tid] = s;
    __syncthreads();
    for (int off = 32; off > 0; off >>= 1) {
        if (tid < off) red[tid] += red[tid + off];
        __syncthreads();
    }
    double nrm = sqrt(red[0]);
    double inv = (nrm > 0.0) ? 1.0 / nrm : 0.0;

    const int o1 = l1 * l1, o2 = l2 * l2, so = l3 * l3;
    for (int idx = tid; idx < tot; idx += 64) {
        int j = idx / (n2 * n3), l = (idx / n3) % n2, n = idx % n3;
        int k = (o1 + j) * 16 + (o2 + l);
        Bt[(size_t)(so + n) * B_STRIDE + k] = (float)(R[idx] * inv);
    }
}

// ------------------------- main WMMA contraction ---------------------------
// One wave per 16-item M-tile.  K = 256 pair-products, N = 32 (two 16 tiles).
// A fragment built on the fly: A[m][i*16+j] = t1[m][i] * t2[m][j].
__global__ void __launch_bounds__(256) cg_wmma_kernel(
    const float* __restrict__ T1, const float* __restrict__ T2,
    const float* __restrict__ Bt, float* __restrict__ Out, int NC) {
    __shared__ float Bs[B_FLOATS];
    for (int i = threadIdx.x; i < B_FLOATS; i += blockDim.x) Bs[i] = Bt[i];
    __syncthreads();

    const int lane   = threadIdx.x & 31;
    const int laneLo = lane & 15;
    const int half   = lane >> 4;          // 0: K slots {0,1}, 1: K slots {2,3}
    const int wave   = blockIdx.x * (blockDim.x >> 5) + (threadIdx.x >> 5);
    const int nWaves = gridDim.x * (blockDim.x >> 5);
    const int nTiles = (NC + 15) >> 4;
    const int row0 = laneLo;               // B row (s) for N-tile 0
    const int row1 = laneLo + 16;          // B row (s) for N-tile 1

    for (int tile = wave; tile < nTiles; tile += nWaves) {
        const int base = tile << 4;
        const int item = base + laneLo;

        float t1r[16];
        float w[8];                        // t2 values for this lane's K slots
        if (item < NC) {
            const float4* p1 = reinterpret_cast<const float4*>(T1 + (size_t)item * 16);
#pragma unroll
            for (int r = 0; r < 4; ++r) {
                float4 v = p1[r];
                t1r[4 * r + 0] = v.x; t1r[4 * r + 1] = v.y;
                t1r[4 * r + 2] = v.z; t1r[4 * r + 3] = v.w;
            }
            const float* p2 = T2 + (size_t)item * 16 + 2 * half;
#pragma unroll
            for (int q = 0; q < 4; ++q) {
                float2 v = *reinterpret_cast<const float2*>(p2 + 4 * q);
                w[2 * q] = v.x; w[2 * q + 1] = v.y;
            }
        } else {
#pragma unroll
            for (int r = 0; r < 16; ++r) t1r[r] = 0.0f;
#pragma unroll
            for (int r = 0; r < 8; ++r) w[r] = 0.0f;
        }

        v8f c0 = {}, c1 = {};
#pragma unroll
        for (int t = 0; t < 64; ++t) {
            const int i = t >> 2;          // t1 feature index (K block of 16)
            const int q = t & 3;           // t2 sub-block
            v2f a;
            a.x = t1r[i] * w[2 * q];
            a.y = t1r[i] * w[2 * q + 1];
            const int k = 4 * t + 2 * half;
            v2f b0 = *reinterpret_cast<const v2f*>(&Bs[row0 * B_STRIDE + k]);
            v2f b1 = *reinterpret_cast<const v2f*>(&Bs[row1 * B_STRIDE + k]);
            c0 = __builtin_amdgcn_wmma_f32_16x16x4_f32(false, a, false, b0,
                                                       (short)0, c0, false, false);
            c1 = __builtin_amdgcn_wmma_f32_16x16x4_f32(false, a, false, b1,
                                                       (short)0, c1, false, false);
        }

        // D layout: VGPR r holds rows {r, r+8}; this lane has rows half*8+r,
        // column laneLo (tile0) / laneLo+16 (tile1).  Only s < 25 is real.
#pragma unroll
        for (int r = 0; r < 8; ++r) {
            const int orow = base + half * 8 + r;
            if (orow < NC) {
                Out[(size_t)orow * 25 + laneLo] = c0[r];
                if (laneLo < 9)
                    Out[(size_t)orow * 25 + 16 + laneLo] = c1[r];
            }
        }
    }
}

// ---------------------------------------------------------------------------
extern "C" void kernel_launch(void* const* d_in, const int* in_sizes, int n_in,
                              void* d_out, int out_size, void* d_ws, size_t ws_size,
                              hipStream_t stream) {
    const float* t1 = (const float*)d_in[0];
    const float* t2 = (const float*)d_in[1];
    float* out = (float*)d_out;
    float* Bt = (float*)d_ws;
    const int NC = in_sizes[0] / 16;       // 50000*64 = 3,200,000

    // build padded Wigner coefficient table in scratch (zero first: padding +
    // uncovered (k,s) cells must be 0).  All async on `stream`: capture-safe.
    hipMemsetAsync(d_ws, 0, (size_t)B_FLOATS * sizeof(float), stream);
    hipLaunchKernelGGL(w3j_gen_kernel, dim3(27), dim3(64), 0, stream, Bt);

    const int blocks = 2048;               // 8 waves/block, ~12 tiles per wave
    hipLaunchKernelGGL(cg_wmma_kernel, dim3(blocks), dim3(256), 0, stream,
                       t1, t2, Bt, out, NC);
}